// VectorQuantizeEMA_36309653520519
// MI455X (gfx1250) — compile-verified
//
#include <hip/hip_runtime.h>
#include <hip/hip_bf16.h>

typedef float v2f __attribute__((ext_vector_type(2)));
typedef float v8f __attribute__((ext_vector_type(8)));

#define DIM        128
#define KCODES     1024
#define NROWS      65536          // 16384*512/128
#define ROWS_PER_WAVE  16
#define WAVES_PER_BLOCK 8
#define ROWS_PER_BLOCK (ROWS_PER_WAVE*WAVES_PER_BLOCK)   // 128
#define NBLOCKS    (NROWS/ROWS_PER_BLOCK)                // 512
#define Z_ELEMS    8388608        // 16384*512
#define N_ELEMS_F  8388608.0f

// ---------------- init: zero histogram, precompute ||e_k||^2 ----------------
__global__ __launch_bounds__(256) void vq_init(const float* __restrict__ E,
                                               float* __restrict__ counts,
                                               float* __restrict__ e2) {
    int k = blockIdx.x * blockDim.x + threadIdx.x;
    if (k < KCODES) {
        counts[k] = 0.0f;
        const float* row = E + (size_t)k * DIM;
        float s = 0.0f;
        #pragma unroll 8
        for (int i = 0; i < DIM; ++i) s += row[i] * row[i];
        e2[k] = s;
    }
}

// ---------------- main: WMMA distance GEMM + argmin + gather ----------------
__global__ __launch_bounds__(256) void vq_main(const float* __restrict__ x,
                                               const float* __restrict__ E,
                                               const float* __restrict__ e2,
                                               float* __restrict__ z,
                                               float* __restrict__ codes,
                                               float* __restrict__ counts,
                                               float* __restrict__ diffPartial) {
    const int lane = threadIdx.x & 31;
    const int wave = threadIdx.x >> 5;
    const int rowBase = blockIdx.x * ROWS_PER_BLOCK + wave * ROWS_PER_WAVE;
    const int m  = lane & 15;     // row (A) / col (B,C) within tile
    const int hi = lane >> 4;     // k-pair selector for 16x16x4 f32 layout

    // Preload A fragments: a[s] covers k = {4s+2*hi, 4s+2*hi+1} of row (rowBase+m)
    v2f a[32];
    const float* xrow = x + (size_t)(rowBase + m) * DIM;
    #pragma unroll
    for (int s = 0; s < 32; ++s)
        a[s] = *(const v2f*)(xrow + 4 * s + 2 * hi);

    // ||x_m||^2 : this lane holds half the row, partner lane (^16) the other half
    float x2p = 0.0f;
    #pragma unroll
    for (int s = 0; s < 32; ++s) x2p += a[s].x * a[s].x + a[s].y * a[s].y;
    float x2 = x2p + __shfl_xor(x2p, 16, 32);

    float bestS[8];
    int   bestI[8];
    #pragma unroll
    for (int j = 0; j < 8; ++j) { bestS[j] = 3.4e38f; bestI[j] = 0; }

    for (int t = 0; t < 64; ++t) {
        const int n = t * 16 + m;                 // codebook column this lane owns
        const float* erow = E + (size_t)n * DIM;
        v8f acc0 = {0,0,0,0,0,0,0,0}, acc1 = acc0, acc2 = acc0, acc3 = acc0;
        #pragma unroll
        for (int s = 0; s < 32; s += 4) {
            v2f b0 = *(const v2f*)(erow + 4 * (s + 0) + 2 * hi);
            v2f b1 = *(const v2f*)(erow + 4 * (s + 1) + 2 * hi);
            v2f b2 = *(const v2f*)(erow + 4 * (s + 2) + 2 * hi);
            v2f b3 = *(const v2f*)(erow + 4 * (s + 3) + 2 * hi);
            acc0 = __builtin_amdgcn_wmma_f32_16x16x4_f32(false, a[s+0], false, b0, (short)0, acc0, false, false);
            acc1 = __builtin_amdgcn_wmma_f32_16x16x4_f32(false, a[s+1], false, b1, (short)0, acc1, false, false);
            acc2 = __builtin_amdgcn_wmma_f32_16x16x4_f32(false, a[s+2], false, b2, (short)0, acc2, false, false);
            acc3 = __builtin_amdgcn_wmma_f32_16x16x4_f32(false, a[s+3], false, b3, (short)0, acc3, false, false);
        }
        const float en2 = e2[n];
        #pragma unroll
        for (int j = 0; j < 8; ++j) {
            float dot = acc0[j] + acc1[j] + acc2[j] + acc3[j];
            float v = en2 - 2.0f * dot;           // dist minus constant ||x||^2
            if (v < bestS[j]) { bestS[j] = v; bestI[j] = n; }  // strict < keeps first min
        }
    }

    // Reduce across the 16 columns held by each half-wave; tie -> lower index
    #pragma unroll
    for (int mask = 1; mask <= 8; mask <<= 1) {
        #pragma unroll
        for (int j = 0; j < 8; ++j) {
            float os = __shfl_xor(bestS[j], mask, 32);
            int   oi = __shfl_xor(bestI[j], mask, 32);
            if (os < bestS[j] || (os == bestS[j] && oi < bestI[j])) {
                bestS[j] = os; bestI[j] = oi;
            }
        }
    }

    __shared__ int   sIdx[WAVES_PER_BLOCK][16];
    __shared__ float sS  [WAVES_PER_BLOCK][16];
    __shared__ float sX2 [WAVES_PER_BLOCK][16];
    __shared__ float sDiff[WAVES_PER_BLOCK];
    if (lane == 0 || lane == 16) {
        #pragma unroll
        for (int j = 0; j < 8; ++j) {
            sIdx[wave][hi * 8 + j] = bestI[j];
            sS  [wave][hi * 8 + j] = bestS[j];
        }
    }
    if (lane < 16) sX2[wave][m] = x2;
    __syncthreads();

    // codes + histogram (lanes 0..15 handle one row each)
    if (lane < 16) {
        int idx = sIdx[wave][lane];
        codes[rowBase + lane] = (float)idx;
        atomicAdd(&counts[idx], 1.0f);            // integer-valued -> exact/deterministic
    }

    // per-row squared error = ||x||^2 + bestS  (== ||x - e_best||^2)
    float dr = (lane < 16) ? (sX2[wave][lane] + sS[wave][lane]) : 0.0f;
    #pragma unroll
    for (int mask = 1; mask <= 16; mask <<= 1) dr += __shfl_xor(dr, mask, 32);
    if (lane == 0) sDiff[wave] = dr;
    __syncthreads();
    if (wave == 0 && lane == 0) {                 // fixed-order sum: deterministic
        float s = 0.0f;
        #pragma unroll
        for (int w = 0; w < WAVES_PER_BLOCK; ++w) s += sDiff[w];
        diffPartial[blockIdx.x] = s;
    }

    // gather: z row <- embedding[bestIdx] (32 lanes x float4 = 128 floats/row)
    #pragma unroll 4
    for (int r = 0; r < ROWS_PER_WAVE; ++r) {
        int idx = sIdx[wave][r];
        const float4 ev = *(const float4*)(E + (size_t)idx * DIM + lane * 4);
        *(float4*)(z + (size_t)(rowBase + r) * DIM + lane * 4) = ev;
    }
}

// ---------------- finalize: diff mean + perplexity ----------------
__global__ __launch_bounds__(256) void vq_finalize(const float* __restrict__ counts,
                                                   const float* __restrict__ diffPartial,
                                                   float* __restrict__ out_diff,
                                                   float* __restrict__ out_ppl) {
    __shared__ float red[256];
    int tid = threadIdx.x;
    if (tid == 0) {
        float s = 0.0f;
        for (int i = 0; i < NBLOCKS; ++i) s += diffPartial[i];   // fixed order
        *out_diff = s / N_ELEMS_F;
    }
    float acc = 0.0f;
    for (int k = tid; k < KCODES; k += 256) {
        float p = counts[k] * (1.0f / (float)NROWS);
        acc += p * logf(p + 1e-5f);
    }
    red[tid] = acc;
    __syncthreads();
    for (int s2 = 128; s2 > 0; s2 >>= 1) {
        if (tid < s2) red[tid] += red[tid + s2];
        __syncthreads();
    }
    if (tid == 0) *out_ppl = -red[0];
}

extern "C" void kernel_launch(void* const* d_in, const int* in_sizes, int n_in,
                              void* d_out, int out_size, void* d_ws, size_t ws_size,
                              hipStream_t stream) {
    const float* x = (const float*)d_in[0];    // [16384, 512]
    const float* E = (const float*)d_in[1];    // [1024, 128]
    float* out   = (float*)d_out;
    float* z     = out;                        // [8388608]
    float* diff  = out + Z_ELEMS;              // [1]
    float* codes = out + Z_ELEMS + 1;          // [65536]
    float* ppl   = out + Z_ELEMS + 1 + NROWS;  // [1]

    float* ws          = (float*)d_ws;
    float* counts      = ws;                   // [1024]
    float* e2          = ws + KCODES;          // [1024]
    float* diffPartial = ws + 2 * KCODES;      // [512]

    vq_init    <<<(KCODES + 255) / 256, 256, 0, stream>>>(E, counts, e2);
    vq_main    <<<NBLOCKS, 256, 0, stream>>>(x, E, e2, z, codes, counts, diffPartial);
    vq_finalize<<<1, 256, 0, stream>>>(counts, diffPartial, diff, ppl);
}